// ChemBERTaMSLoss_67680094650559
// MI455X (gfx1250) — compile-verified
//
#include <hip/hip_runtime.h>
#include <math.h>

// ---------------------------------------------------------------------------
// ChemBERTa MS loss for MI455X (gfx1250, wave32).
// Memory-bound: ~396MB read -> ~17us floor @ 23.3 TB/s. One pass per array.
// ---------------------------------------------------------------------------

#define BATCH   8192
#define MZ      2000
#define NF      2048
#define EPSF    1.19209290e-07f

typedef float v2f __attribute__((ext_vector_type(2)));
typedef float v8f __attribute__((ext_vector_type(8)));

__device__ __forceinline__ float bce_logits(float x, float y) {
    // max(x,0) - x*y + log1p(exp(-|x|)), hardware trans ops
    float e = __expf(-fabsf(x));
    return fmaxf(x, 0.f) - x * y + __logf(1.f + e);
}

__device__ __forceinline__ float sigmoidf_fast(float x) {
    return 1.f / (1.f + __expf(-x));
}

// ---------------------------------------------------------------------------
// Kernel A: one block (256 threads) per row. Fuses:
//   - weighted BCE partial (prob loss)
//   - masked MSE partials (intensity loss)
//   - per-row EMD (row sums -> normalize -> block scan -> |cdf| sum)
// Each thread owns 8 contiguous elements (2x float4 loads). 250*8 = 2000.
// ---------------------------------------------------------------------------
__global__ __launch_bounds__(256) void row_kernel(
    const float* __restrict__ PI, const float* __restrict__ PL,
    const float* __restrict__ TI, const float* __restrict__ TP,
    float* __restrict__ probP, float* __restrict__ intNP,
    float* __restrict__ intDP, float* __restrict__ wsNumP,
    float* __restrict__ wsValP)
{
    __shared__ float s_scan[256];
    __shared__ float s_w[32];
    __shared__ float s_b[2];

    const int row  = blockIdx.x;
    const int t    = threadIdx.x;
    const int lane = t & 31;
    const int wid  = t >> 5;
    const size_t base = (size_t)row * MZ;
    const bool active = (t < MZ / 8);   // 250 active threads

    float d[8], u[8];
    float pv[8], lv[8], tv[8], qv[8];
#pragma unroll
    for (int j = 0; j < 8; ++j) { d[j] = 0.f; u[j] = 0.f; }

    float probAcc = 0.f, intN = 0.f, intD = 0.f, sd = 0.f, su = 0.f;

    if (active) {
        const size_t off = base + (size_t)t * 8;
        const float4 a0 = *(const float4*)(PI + off);
        const float4 a1 = *(const float4*)(PI + off + 4);
        const float4 b0 = *(const float4*)(PL + off);
        const float4 b1 = *(const float4*)(PL + off + 4);
        const float4 c0 = *(const float4*)(TI + off);
        const float4 c1 = *(const float4*)(TI + off + 4);
        const float4 e0 = *(const float4*)(TP + off);
        const float4 e1 = *(const float4*)(TP + off + 4);
        pv[0]=a0.x; pv[1]=a0.y; pv[2]=a0.z; pv[3]=a0.w; pv[4]=a1.x; pv[5]=a1.y; pv[6]=a1.z; pv[7]=a1.w;
        lv[0]=b0.x; lv[1]=b0.y; lv[2]=b0.z; lv[3]=b0.w; lv[4]=b1.x; lv[5]=b1.y; lv[6]=b1.z; lv[7]=b1.w;
        tv[0]=c0.x; tv[1]=c0.y; tv[2]=c0.z; tv[3]=c0.w; tv[4]=c1.x; tv[5]=c1.y; tv[6]=c1.z; tv[7]=c1.w;
        qv[0]=e0.x; qv[1]=e0.y; qv[2]=e0.z; qv[3]=e0.w; qv[4]=e1.x; qv[5]=e1.y; qv[6]=e1.z; qv[7]=e1.w;
#pragma unroll
        for (int j = 0; j < 8; ++j) {
            const float x = lv[j], y = qv[j];
            const float bce = bce_logits(x, y);
            const bool  pos = (y > 0.5f);
            probAcc += bce * (pos ? 5.0f : 1.0f);
            const float iw = pos ? 1.0f : 0.0f;
            const float df = pv[j] - tv[j];
            intN += iw * df * df;
            intD += iw;
            const float dv = fmaxf(pv[j], 0.f) * sigmoidf_fast(x);
            const float uv = tv[j] * y;
            d[j] = dv; u[j] = uv;
            sd += dv; su += uv;
        }
    }

    // ---- block reduce (sd, su) with broadcast ----
    {
        float x = sd, y = su;
#pragma unroll
        for (int m = 16; m >= 1; m >>= 1) {
            x += __shfl_xor(x, m, 32);
            y += __shfl_xor(y, m, 32);
        }
        if (lane == 0) { s_w[wid] = x; s_w[8 + wid] = y; }
    }
    __syncthreads();
    if (t == 0) {
        float ax = 0.f, ay = 0.f;
        for (int i = 0; i < 8; ++i) { ax += s_w[i]; ay += s_w[8 + i]; }
        s_b[0] = ax; s_b[1] = ay;
    }
    __syncthreads();

    const float Sp = s_b[0], Sq = s_b[1];
    // double normalization exactly as the reference applies it
    const float psum1 = Sp / (Sp + EPSF);
    const float qsum1 = Sq / (Sq + EPSF);
    const float rNp = 1.f / ((Sp + EPSF) * (psum1 + EPSF));
    const float rNq = 1.f / ((Sq + EPSF) * (qsum1 + EPSF));
    const float psum = psum1 / (psum1 + EPSF);
    const float qsum = qsum1 / (qsum1 + EPSF);

    // per-thread inclusive prefix of v = p - q
    float pref[8];
    float run = 0.f;
#pragma unroll
    for (int j = 0; j < 8; ++j) {
        const float v = d[j] * rNp - u[j] * rNq;
        run += v;
        pref[j] = run;
    }

    // ---- block inclusive scan of thread totals (Hillis-Steele, 256 wide) ----
    s_scan[t] = run;
    __syncthreads();
#pragma unroll
    for (int off = 1; off < 256; off <<= 1) {
        float add = 0.f;
        if (t >= off) add = s_scan[t - off];
        __syncthreads();
        s_scan[t] += add;
        __syncthreads();
    }
    const float excl = (t == 0) ? 0.f : s_scan[t - 1];

    float absAcc = 0.f;
    if (active) {
        const int gbase = t * 8;
#pragma unroll
        for (int j = 0; j < 8; ++j) {
            const float c = excl + pref[j];
            if (gbase + j < MZ - 1) absAcc += fabsf(c);  // cdf_diff[:, :-1]
        }
    }

    // ---- block reduce (probAcc, intN, intD, absAcc) ----
    {
        float a0 = probAcc, a1 = intN, a2 = intD, a3 = absAcc;
#pragma unroll
        for (int m = 16; m >= 1; m >>= 1) {
            a0 += __shfl_xor(a0, m, 32);
            a1 += __shfl_xor(a1, m, 32);
            a2 += __shfl_xor(a2, m, 32);
            a3 += __shfl_xor(a3, m, 32);
        }
        if (lane == 0) {
            s_w[wid] = a0; s_w[8 + wid] = a1; s_w[16 + wid] = a2; s_w[24 + wid] = a3;
        }
    }
    __syncthreads();
    if (t == 0) {
        float A0 = 0.f, A1 = 0.f, A2 = 0.f, A3 = 0.f;
        for (int i = 0; i < 8; ++i) {
            A0 += s_w[i]; A1 += s_w[8 + i]; A2 += s_w[16 + i]; A3 += s_w[24 + i];
        }
        probP[row] = A0;
        intNP[row] = A1;
        intDP[row] = A2;
        const float scale = 1.0f / ((float)(MZ - 1) + EPSF);
        const float w1    = A3 * scale;
        const float valid = (qsum > EPSF) ? 1.f : 0.f;
        const float per   = (psum > EPSF) ? w1 : 1.f;
        wsNumP[row] = per * valid;
        wsValP[row] = valid;
    }
}

// ---------------------------------------------------------------------------
// Kernel B: morgan BCE partial sums. Grid-stride over 4M float4/int4 vectors.
// ---------------------------------------------------------------------------
#define MORGAN_BLOCKS 1024
__global__ __launch_bounds__(256) void morgan_kernel(
    const float* __restrict__ X, const int* __restrict__ Y,
    float* __restrict__ morganP)
{
    __shared__ float s_w[8];
    const int t = threadIdx.x, lane = t & 31, wid = t >> 5;
    const int nvec = (BATCH * NF) / 4;          // 4,194,304
    const float4* x4 = (const float4*)X;
    const int4*   y4 = (const int4*)Y;
    float acc = 0.f;
    for (int i = blockIdx.x * 256 + t; i < nvec; i += MORGAN_BLOCKS * 256) {
        const float4 xv = x4[i];
        const int4   yv = y4[i];
        acc += bce_logits(xv.x, (float)yv.x);
        acc += bce_logits(xv.y, (float)yv.y);
        acc += bce_logits(xv.z, (float)yv.z);
        acc += bce_logits(xv.w, (float)yv.w);
    }
#pragma unroll
    for (int m = 16; m >= 1; m >>= 1) acc += __shfl_xor(acc, m, 32);
    if (lane == 0) s_w[wid] = acc;
    __syncthreads();
    if (t == 0) {
        float a = 0.f;
        for (int i = 0; i < 8; ++i) a += s_w[i];
        morganP[blockIdx.x] = a;
    }
}

// ---------------------------------------------------------------------------
// Kernel C: deterministic final combine, single wave32.
// Cross-lane sums via V_WMMA_F32_16X16X4_F32 with an all-ones A matrix:
// D[m][n] = sum_k B[k][n]  ->  lanes n and n+16 folded in one matrix op
// (robust to exact K-row->lane-half mapping since A is all ones),
// then 4 shfl_xor steps finish the 16-way reduction.
// ---------------------------------------------------------------------------
__device__ __forceinline__ float wave_sum_wmma(float x) {
    v2f a; a[0] = 1.f; a[1] = 1.f;       // A = ones(16x4)
    v2f b; b[0] = x;   b[1] = 0.f;       // B: one value per lane, rest zero
    v8f c = {};
    c = __builtin_amdgcn_wmma_f32_16x16x4_f32(
        /*neg_a=*/false, a, /*neg_b=*/false, b,
        /*c_mod=*/(short)0, c, /*reuse_a=*/false, /*reuse_b=*/false);
    float s = c[0];                       // colsum[lane & 15] in every lane
    s += __shfl_xor(s, 8, 32);
    s += __shfl_xor(s, 4, 32);
    s += __shfl_xor(s, 2, 32);
    s += __shfl_xor(s, 1, 32);
    return s;                             // total in all lanes
}

__global__ __launch_bounds__(32) void final_kernel(
    const float* __restrict__ W, float* __restrict__ out)
{
    const int lane = threadIdx.x;
    float sProb = 0.f, sIN = 0.f, sID = 0.f, sWN = 0.f, sWV = 0.f, sM = 0.f;
    for (int i = lane; i < BATCH; i += 32) {
        sProb += W[i];
        sIN   += W[BATCH + i];
        sID   += W[2 * BATCH + i];
        sWN   += W[3 * BATCH + i];
        sWV   += W[4 * BATCH + i];
    }
    for (int i = lane; i < MORGAN_BLOCKS; i += 32) sM += W[5 * BATCH + i];

    sProb = wave_sum_wmma(sProb);
    sIN   = wave_sum_wmma(sIN);
    sID   = wave_sum_wmma(sID);
    sWN   = wave_sum_wmma(sWN);
    sWV   = wave_sum_wmma(sWV);
    sM    = wave_sum_wmma(sM);

    if (lane == 0) {
        const float prob_loss      = sProb / (float)((long)BATCH * MZ);
        const float intensity_loss = sIN / (sID + EPSF);
        const float morgan_loss    = sM / (float)((long)BATCH * NF);
        const float ws_loss = (sWV > 0.f) ? (sWN / fmaxf(sWV, 1.f)) : 0.f;
        float total = 0.3f * prob_loss + 0.1f * intensity_loss
                    + 0.5f * ws_loss  + 0.1f * morgan_loss;
        if (isnan(total) || isinf(total)) total = 100000.0f;
        out[0] = total;
    }
}

// ---------------------------------------------------------------------------
extern "C" void kernel_launch(void* const* d_in, const int* in_sizes, int n_in,
                              void* d_out, int out_size, void* d_ws, size_t ws_size,
                              hipStream_t stream) {
    (void)in_sizes; (void)n_in; (void)out_size; (void)ws_size;
    const float* PI = (const float*)d_in[0];   // pred_intensities   [B,MZ]
    const float* PL = (const float*)d_in[1];   // pred_probs_logits  [B,MZ]
    const float* PM = (const float*)d_in[2];   // pred_morgan_logits [B,NF]
    const float* TI = (const float*)d_in[3];   // true_intensities   [B,MZ]
    const float* TP = (const float*)d_in[4];   // true_probs         [B,MZ]
    const int*   TM = (const int*)d_in[5];     // true_morgan        [B,NF]

    float* W = (float*)d_ws;
    float* probP  = W;                  // [8192]
    float* intNP  = W + BATCH;          // [8192]
    float* intDP  = W + 2 * BATCH;      // [8192]
    float* wsNumP = W + 3 * BATCH;      // [8192]
    float* wsValP = W + 4 * BATCH;      // [8192]
    float* morgP  = W + 5 * BATCH;      // [1024]

    row_kernel<<<BATCH, 256, 0, stream>>>(PI, PL, TI, TP,
                                          probP, intNP, intDP, wsNumP, wsValP);
    morgan_kernel<<<MORGAN_BLOCKS, 256, 0, stream>>>(PM, TM, morgP);
    final_kernel<<<1, 32, 0, stream>>>(W, (float*)d_out);
}